// ContextualViewModel_62191126446676
// MI455X (gfx1250) — compile-verified
//
#include <hip/hip_runtime.h>
#include <hip/hip_bf16.h>

typedef __attribute__((ext_vector_type(2))) float v2f;
typedef __attribute__((ext_vector_type(8))) float v8f;

#define FEAT 7

// ---------------------------------------------------------------------------
// WMMA kernel: one wave handles 4 tiles of 16 pixels (64 pixels / wave).
// D(16x16) = A(16x4 pixels x f) * B(4x16 f x g), K=8 via two 16x16x4 f32
// WMMAs. d[i,j] is folded into the A rows before the matmul (d*(x@W)=(d*x)@W).
// Fully branch-free loads; only the store clause is lane-masked (n < 7).
// ---------------------------------------------------------------------------
__global__ __launch_bounds__(256) void cvm_wmma_kernel(
    const float* __restrict__ x,   // (P, 7)
    const float* __restrict__ d,   // (P,)
    const float* __restrict__ W,   // (7, 7) row-major: W[f*7+g]
    float* __restrict__ out)       // (P, 7)
{
  const int lane = threadIdx.x & 31;
  const int wave = threadIdx.x >> 5;
  const int half = lane >> 4;        // 0: lanes 0-15, 1: lanes 16-31
  const int n    = lane & 15;        // N (out feature) for B/C/D, M (pixel) for A

  const long waveId    = (long)blockIdx.x * 8 + wave;   // 8 waves / 256-thr block
  const long pixelBase = waveId * 64;

  // --- Build B operands from W once per wave (branch-free) ----------------
  // 32-bit B 4x16 layout (mirrors A): lanes 0-15: v0=K0, v1=K1;
  //                                   lanes 16-31: v0=K2, v1=K3.
  v2f b0, b1;
  {
    const int   f0 = 2 * half;                   // 0 / 2
    const int   f1 = 4 + 2 * half;               // 4 / 6
    const int   nc = (n < FEAT) ? n : (FEAT - 1);
    const float ns = (n < FEAT) ? 1.0f : 0.0f;
    b0.x = W[(f0    ) * FEAT + nc] * ns;                    // f = 0 / 2
    b0.y = W[(f0 + 1) * FEAT + nc] * ns;                    // f = 1 / 3
    b1.x = W[(f1    ) * FEAT + nc] * ns;                    // f = 4 / 6
    const int f1y = (half == 0) ? (f1 + 1) : (FEAT - 1);    // f = 5 ; clamp f=7
    b1.y = W[f1y * FEAT + nc] * ((half == 0) ? ns : 0.0f);  // f = 5 ; f=7 -> 0
  }

  // Per-lane base pointers; per-tile addressing is immediate offsets only.
  const float* xq = x + (pixelBase + n) * FEAT + 2 * half; // A-row f base
  const float* dq = d + pixelBase + n;
  float*       oq = out + (pixelBase + 8 * half) * FEAT + n;
  const bool   storing = (n < FEAT);

#pragma unroll
  for (int t = 0; t < 4; ++t) {
    const float dv = dq[t * 16];

    // A 16x4 f32 layout: lanes 0-15: v0=K0,v1=K1; lanes 16-31: v0=K2,v1=K3.
    // Two b64 loads; the half==1 slot f=7 reads the neighbor pixel's f0 and
    // is zeroed via the 0-scale (launch guarantees this read is in-bounds).
    v2f a0, a1;
    a0.x = xq[t * 112    ];          // f = 0 / 2
    a0.y = xq[t * 112 + 1];          // f = 1 / 3
    a1.x = xq[t * 112 + 4];          // f = 4 / 6
    a1.y = xq[t * 112 + 5];          // f = 5 / (7 -> zeroed)
    const float dhi = (half == 0) ? dv : 0.0f;
    a0.x *= dv;  a0.y *= dv;
    a1.x *= dv;  a1.y *= dhi;

    v8f c = {};
#if __has_builtin(__builtin_amdgcn_wmma_f32_16x16x4_f32)
    c = __builtin_amdgcn_wmma_f32_16x16x4_f32(false, a0, false, b0,
                                              (short)0, c, false, false);
    c = __builtin_amdgcn_wmma_f32_16x16x4_f32(false, a1, false, b1,
                                              (short)0, c, false, false);
#else
    #pragma unroll
    for (int r = 0; r < 8; ++r) {
      const long pr = pixelBase + t * 16 + r + 8 * half;
      float acc = 0.f;
      #pragma unroll
      for (int f = 0; f < FEAT; ++f) acc += x[pr * FEAT + f] * W[f * FEAT + ((n < FEAT) ? n : 0)];
      c[r] = acc * d[pr];
    }
#endif

    // D layout: VGPR r, lanes 0-15 -> (M=r, N=n); lanes 16-31 -> (M=r+8, N=n).
    if (storing) {
      #pragma unroll
      for (int r = 0; r < 8; ++r)
        __builtin_nontemporal_store(c[r], oq + t * 112 + r * FEAT);
    }
  }
}

// ---------------------------------------------------------------------------
// Scalar tail kernel: handles the last (nPix - mainPix) pixels. Always covers
// at least the final pixel so the WMMA kernel's speculative f=7 read stays
// inside x.
// ---------------------------------------------------------------------------
__global__ void cvm_tail_kernel(const float* __restrict__ x,
                                const float* __restrict__ d,
                                const float* __restrict__ W,
                                float* __restrict__ out,
                                long start, long nPix)
{
  long p = start + (long)blockIdx.x * blockDim.x + threadIdx.x;
  if (p >= nPix) return;
  const float* xp = x + p * FEAT;
  float xr[FEAT];
  #pragma unroll
  for (int f = 0; f < FEAT; ++f) xr[f] = xp[f];
  const float dv = d[p];
  #pragma unroll
  for (int g = 0; g < FEAT; ++g) {
    float acc = 0.f;
    #pragma unroll
    for (int f = 0; f < FEAT; ++f) acc += xr[f] * W[f * FEAT + g];
    out[p * FEAT + g] = dv * acc;
  }
}

extern "C" void kernel_launch(void* const* d_in, const int* in_sizes, int n_in,
                              void* d_out, int out_size, void* d_ws, size_t ws_size,
                              hipStream_t stream)
{
  const float* x = (const float*)d_in[0];  // (H, W, 7)
  const float* d = (const float*)d_in[1];  // (H, W)
  const float* W = (const float*)d_in[2];  // (7, 7)
  float* out = (float*)d_out;              // (H, W, 7)

  const long nPix = (long)in_sizes[1];     // H * W
  // Reserve at least one pixel for the tail so the WMMA kernel's speculative
  // +1-element read never leaves the x allocation.
  const long mainPix = (nPix > 0) ? ((nPix - 1) / 64) * 64 : 0;
  const long blocks  = mainPix / (64 * 8); // 8 waves (256 thr) per block

  if (blocks > 0)
    cvm_wmma_kernel<<<(int)blocks, 256, 0, stream>>>(x, d, W, out);

  const long rem = nPix - mainPix;
  if (rem > 0)
    cvm_tail_kernel<<<(int)((rem + 255) / 256), 256, 0, stream>>>(
        x, d, W, out, mainPix, nPix);
}